// GraphSAGE_15547781611787
// MI455X (gfx1250) — compile-verified
//
#include <hip/hip_runtime.h>

// GCN 3-layer forward for MI455X (gfx1250, wave32).
// GEMM via V_WMMA_F32_16X16X4_F32 (fp32, K=64 as 16 chained k=4 steps).
// Edge aggregation via per-float GLOBAL_ATOMIC_ADD_F32 into an L2-resident
// accumulator (feature arrays = 25.6MB << 192MB L2). Self-loop term folded
// into the elementwise bias/ReLU pass (no atomics needed for it).

typedef __attribute__((ext_vector_type(2))) float v2f;
typedef __attribute__((ext_vector_type(8))) float v8f;

#define FDIM 64

// ---------- utility ----------
__global__ void fill_f32(float* __restrict__ p, long long n, float v) {
  long long i = (long long)blockIdx.x * blockDim.x + threadIdx.x;
  if (i < n) p[i] = v;
}

// deg[i] starts at 1.0 (self-loop weight); add 1 per non-self out-edge.
__global__ void deg_accum(const int* __restrict__ src, const int* __restrict__ dst,
                          float* __restrict__ deg, int E) {
  int e = blockIdx.x * blockDim.x + threadIdx.x;
  if (e >= E) return;
  int s = src[e], d = dst[e];
  if (s != d) atomicAdd(&deg[s], 1.0f);
}

__global__ void deg_to_dinv(float* __restrict__ dinv, int N) {
  int i = blockIdx.x * blockDim.x + threadIdx.x;
  if (i >= N) return;
  float v = dinv[i];
  dinv[i] = (v > 0.0f) ? rsqrtf(v) : 0.0f;
}

__global__ void norm_kernel(const int* __restrict__ src, const int* __restrict__ dst,
                            const float* __restrict__ dinv, float* __restrict__ norm, int E) {
  int e = blockIdx.x * blockDim.x + threadIdx.x;
  if (e >= E) return;
  int s = src[e], d = dst[e];
  norm[e] = (s != d) ? (dinv[s] * dinv[d]) : 0.0f;  // ew=0 for src==dst edges
}

// ---------- WMMA fp32 GEMM: out[N,64] = A[N,64] @ W[64,64] ----------
// One wave per 16x16 output tile, 8 waves per block.
__global__ __launch_bounds__(256) void gemm64_wmma(const float* __restrict__ A,
                                                   const float* __restrict__ W,
                                                   float* __restrict__ out,
                                                   int nrows, int rowTiles) {
  int wave  = blockIdx.x * 8 + (threadIdx.x >> 5);   // wave-uniform
  int lane  = threadIdx.x & 31;
  int tileM = wave >> 2;     // 4 column tiles (64/16)
  int tileN = wave & 3;
  if (tileM >= rowTiles) return;                     // wave-uniform exit: EXEC stays full

  int m0 = tileM << 4;
  int n0 = tileN << 4;
  int lh = lane & 15;        // lane within half
  int hi = lane >> 4;        // which half (carries K split)

  // A fragment source row (clamped for generality; N=100000 is 16-divisible)
  int arow = m0 + lh; if (arow >= nrows) arow = nrows - 1;
  const float* Ap = A + (long long)arow * FDIM;
  const float* Wp = W + n0 + lh;

  v8f c = {};
#pragma unroll
  for (int kk = 0; kk < 16; ++kk) {
    int kb = (kk << 2) + (hi << 1);     // K base for this lane-half
    v2f a, b;
    a.x = Ap[kb];                       // A[m][kb], A[m][kb+1] -> global_load_b64
    a.y = Ap[kb + 1];
    b.x = Wp[(long long)kb * FDIM];     // W[kb][n], W[kb+1][n]
    b.y = Wp[(long long)(kb + 1) * FDIM];
    c = __builtin_amdgcn_wmma_f32_16x16x4_f32(
        /*neg_a=*/false, a, /*neg_b=*/false, b,
        /*c_mod=*/(short)0, c, /*reuse_a=*/false, /*reuse_b=*/false);
  }

#pragma unroll
  for (int r = 0; r < 8; ++r) {
    int orow = m0 + (hi << 3) + r;      // D layout: VGPR r holds rows r / r+8
    if (orow < nrows) out[(long long)orow * FDIM + n0 + lh] = c[r];
  }
}

// ---------- edge aggregation: agg[src] += norm * xw[dst] ----------
// 16 threads per edge, float4 each -> coalesced 256B gather per edge.
__global__ void edge_aggregate(const int* __restrict__ src, const int* __restrict__ dst,
                               const float* __restrict__ norm, const float* __restrict__ xw,
                               float* __restrict__ agg, int E) {
  long long t = (long long)blockIdx.x * blockDim.x + threadIdx.x;
  long long e = t >> 4;
  if (e >= E) return;
  float w = norm[e];
  if (w == 0.0f) return;                // self-dup edges contribute nothing
  int c = (int)(t & 15) << 2;
  int s = src[e], d = dst[e];
  float4 v = *(const float4*)(xw + (long long)d * FDIM + c);
  float* o = agg + (long long)s * FDIM + c;
  atomicAdd(o + 0, w * v.x);
  atomicAdd(o + 1, w * v.y);
  atomicAdd(o + 2, w * v.z);
  atomicAdd(o + 3, w * v.w);
}

// ---------- h = [relu](agg + dinv^2 * xw + b)  (self-loop folded in) ----------
__global__ void bias_act(const float* __restrict__ agg, const float* __restrict__ xw,
                         const float* __restrict__ dinv, const float* __restrict__ b,
                         float* __restrict__ h, long long total, int do_relu) {
  long long t = (long long)blockIdx.x * blockDim.x + threadIdx.x;
  if (t >= total) return;
  int i = (int)(t >> 6);
  int f = (int)(t & 63);
  float di = dinv[i];
  float v = agg[t] + di * di * xw[t] + b[f];
  h[t] = do_relu ? fmaxf(v, 0.0f) : v;
}

// ---------- global add pool: out[batch[i]] += h[i] ----------
__global__ void pool_kernel(const float* __restrict__ h, const int* __restrict__ batch,
                            float* __restrict__ out, int N) {
  long long t = (long long)blockIdx.x * blockDim.x + threadIdx.x;
  long long i = t >> 4;
  if (i >= N) return;
  int c = (int)(t & 15) << 2;
  int g = batch[i];
  float4 v = *(const float4*)(h + i * FDIM + c);
  float* o = out + (long long)g * FDIM + c;
  atomicAdd(o + 0, v.x);
  atomicAdd(o + 1, v.y);
  atomicAdd(o + 2, v.z);
  atomicAdd(o + 3, v.w);
}

static inline int cdiv_ll(long long a, long long b) { return (int)((a + b - 1) / b); }

extern "C" void kernel_launch(void* const* d_in, const int* in_sizes, int n_in,
                              void* d_out, int out_size, void* d_ws, size_t ws_size,
                              hipStream_t stream) {
  const int N = in_sizes[0] / FDIM;
  const int E = in_sizes[1] / 2;

  const float* x     = (const float*)d_in[0];
  const int*   src   = (const int*)d_in[1];       // edge_index[0,:]
  const int*   dst   = src + E;                   // edge_index[1,:]
  const int*   batch = (const int*)d_in[2];
  const float* W1 = (const float*)d_in[3];
  const float* b1 = (const float*)d_in[4];
  const float* W2 = (const float*)d_in[5];
  const float* b2 = (const float*)d_in[6];
  const float* W3 = (const float*)d_in[7];
  const float* b3 = (const float*)d_in[8];

  // workspace carve (all offsets 16B aligned: N, E multiples of 4)
  float* dinv = (float*)d_ws;                       // N
  float* norm = dinv + N;                           // E
  float* xw   = norm + E;                           // N*64
  float* hA   = xw + (long long)N * FDIM;           // N*64
  float* hB   = hA + (long long)N * FDIM;           // N*64

  const int TB = 256;
  const long long NF = (long long)N * FDIM;
  const int rowTiles = (N + 15) / 16;
  const int gemmBlocks = (rowTiles * 4 + 7) / 8;

  // graph normalization (computed once, reused 3x)
  fill_f32<<<cdiv_ll(N, TB), TB, 0, stream>>>(dinv, N, 1.0f);          // self-loop weight
  deg_accum<<<cdiv_ll(E, TB), TB, 0, stream>>>(src, dst, dinv, E);
  deg_to_dinv<<<cdiv_ll(N, TB), TB, 0, stream>>>(dinv, N);
  norm_kernel<<<cdiv_ll(E, TB), TB, 0, stream>>>(src, dst, dinv, norm, E);

  // layer 1: x -> hA (ReLU)
  gemm64_wmma<<<gemmBlocks, 256, 0, stream>>>(x, W1, xw, N, rowTiles);
  fill_f32<<<cdiv_ll(NF, TB), TB, 0, stream>>>(hA, NF, 0.0f);
  edge_aggregate<<<cdiv_ll((long long)E * 16, TB), TB, 0, stream>>>(src, dst, norm, xw, hA, E);
  bias_act<<<cdiv_ll(NF, TB), TB, 0, stream>>>(hA, xw, dinv, b1, hA, NF, 1);

  // layer 2: hA -> hB (ReLU)
  gemm64_wmma<<<gemmBlocks, 256, 0, stream>>>(hA, W2, xw, N, rowTiles);
  fill_f32<<<cdiv_ll(NF, TB), TB, 0, stream>>>(hB, NF, 0.0f);
  edge_aggregate<<<cdiv_ll((long long)E * 16, TB), TB, 0, stream>>>(src, dst, norm, xw, hB, E);
  bias_act<<<cdiv_ll(NF, TB), TB, 0, stream>>>(hB, xw, dinv, b2, hB, NF, 1);

  // layer 3: hB -> hA (no ReLU)
  gemm64_wmma<<<gemmBlocks, 256, 0, stream>>>(hB, W3, xw, N, rowTiles);
  fill_f32<<<cdiv_ll(NF, TB), TB, 0, stream>>>(hA, NF, 0.0f);
  edge_aggregate<<<cdiv_ll((long long)E * 16, TB), TB, 0, stream>>>(src, dst, norm, xw, hA, E);
  bias_act<<<cdiv_ll(NF, TB), TB, 0, stream>>>(hA, xw, dinv, b3, hA, NF, 0);

  // global add pool -> d_out [512, 64]
  fill_f32<<<cdiv_ll(out_size, TB), TB, 0, stream>>>((float*)d_out, out_size, 0.0f);
  pool_kernel<<<cdiv_ll((long long)N * 16, TB), TB, 0, stream>>>(hA, batch, (float*)d_out, N);
}